// SSIMLoss_31636729102952
// MI455X (gfx1250) — compile-verified
//
#include <hip/hip_runtime.h>

// SSIM loss (16x3x512x512 f32) for gfx1250.
// Separable 11-tap Gaussian blur expressed as banded-Toeplitz matmuls on the
// f32 WMMA pipe: Out(16x16) = In(16x28) x T(28x16) via 7x V_WMMA_F32_16X16X4_F32
// per pass per quantity (x, y, x^2, y^2, xy). One wave32 per 16x16 output tile.

typedef __attribute__((ext_vector_type(2))) float v2f;
typedef __attribute__((ext_vector_type(8))) float v8f;

#define WIN   11
#define PAD   5
#define TILE  16
#define KPAD  28      // 26-wide halo span padded to a multiple of 4
#define NK    7       // K chunks of 4
#define ROWS  32      // loaded rows: r0-5 .. r0+26 (covers both 16-row H tiles)
#define S_IN  29      // LDS row stride (odd -> fewer bank conflicts)
#define S_H   17
#define IMG_H 512
#define IMG_W 512
#define PLANES 48     // 16 batch * 3 channels

// Normalized 1-D Gaussian, sigma=1.5 (outer product == reference 2-D window).
__constant__ float c_gw[WIN] = {
    0.0010284f, 0.0075987f, 0.0360008f, 0.1093608f, 0.2130056f,
    0.2660117f,
    0.2130056f, 0.1093608f, 0.0360008f, 0.0075987f, 0.0010284f
};

__device__ __forceinline__ float gwv(int i) {
    return (i >= 0 && i < WIN) ? c_gw[i] : 0.0f;
}

__global__ __launch_bounds__(32) void ssim_tile_wmma(
        const float* __restrict__ img1, const float* __restrict__ img2,
        float* __restrict__ target, int atomicMode)
{
    __shared__ float sIn1[ROWS][S_IN];
    __shared__ float sIn2[ROWS][S_IN];
    __shared__ float sH[5][ROWS][S_H];   // horizontally blurred quantities

    const int lane = threadIdx.x;        // 0..31 (wave32)
    const int n15  = lane & 15;
    const int koff = (lane < 16) ? 0 : 2;

    const int c0 = blockIdx.x * TILE;
    const int r0 = blockIdx.y * TILE;
    const size_t planeOff = (size_t)blockIdx.z * IMG_H * IMG_W;

    // ---- Phase 0: cooperative zero-padded load of 32x28 halo tiles ----
    #pragma unroll
    for (int it = 0; it < (ROWS * KPAD) / 32; ++it) {
        int idx = it * 32 + lane;
        int rr = idx / KPAD;
        int cc = idx - rr * KPAD;
        int gr = r0 - PAD + rr;
        int gc = c0 - PAD + cc;
        float a = 0.0f, b = 0.0f;
        if (gr >= 0 && gr < IMG_H && gc >= 0 && gc < IMG_W) {
            size_t off = planeOff + (size_t)gr * IMG_W + (size_t)gc;
            a = img1[off];
            b = img2[off];
        }
        sIn1[rr][cc] = a;
        sIn2[rr][cc] = b;
    }
    __syncthreads();

    // Per-lane Toeplitz operand values. With the assumed B layout (v0: K=4j+koff,
    // v1: K=4j+koff+1, N=lane&15) the horizontal B and vertical A values coincide.
    v2f tp[NK];
    #pragma unroll
    for (int j = 0; j < NK; ++j) {
        tp[j].x = gwv(4 * j + koff - n15);
        tp[j].y = gwv(4 * j + koff + 1 - n15);
    }

    const v8f zero8 = {0.f, 0.f, 0.f, 0.f, 0.f, 0.f, 0.f, 0.f};

    // ---- Phase 1: horizontal blur of x, y, x^2, y^2, xy over 32 halo rows ----
    #pragma unroll
    for (int t = 0; t < 2; ++t) {
        v8f a0 = zero8, a1 = zero8, a2 = zero8, a3 = zero8, a4 = zero8;
        const int row = t * 16 + n15;
        #pragma unroll
        for (int j = 0; j < NK; ++j) {
            const int col = 4 * j + koff;
            float x0 = sIn1[row][col], x1 = sIn1[row][col + 1];
            float y0 = sIn2[row][col], y1 = sIn2[row][col + 1];
            v2f ax  = { x0, x1 };
            v2f ay  = { y0, y1 };
            v2f axx = { x0 * x0, x1 * x1 };
            v2f ayy = { y0 * y0, y1 * y1 };
            v2f axy = { x0 * y0, x1 * y1 };
            v2f b   = tp[j];
            a0 = __builtin_amdgcn_wmma_f32_16x16x4_f32(false, ax,  false, b, (short)0, a0, false, false);
            a1 = __builtin_amdgcn_wmma_f32_16x16x4_f32(false, ay,  false, b, (short)0, a1, false, false);
            a2 = __builtin_amdgcn_wmma_f32_16x16x4_f32(false, axx, false, b, (short)0, a2, false, false);
            a3 = __builtin_amdgcn_wmma_f32_16x16x4_f32(false, ayy, false, b, (short)0, a3, false, false);
            a4 = __builtin_amdgcn_wmma_f32_16x16x4_f32(false, axy, false, b, (short)0, a4, false, false);
        }
        // Spill C tiles to LDS using the documented C/D layout:
        // lane<16: M=v, N=lane ; lane>=16: M=v+8, N=lane-16.
        const int rbase = t * 16 + ((lane < 16) ? 0 : 8);
        #pragma unroll
        for (int v = 0; v < 8; ++v) {
            sH[0][rbase + v][n15] = a0[v];
            sH[1][rbase + v][n15] = a1[v];
            sH[2][rbase + v][n15] = a2[v];
            sH[3][rbase + v][n15] = a3[v];
            sH[4][rbase + v][n15] = a4[v];
        }
    }
    __syncthreads();

    // ---- Phase 2: vertical blur: Out = Tv(16x28) x H(28x16) ----
    v8f o0 = zero8, o1 = zero8, o2 = zero8, o3 = zero8, o4 = zero8;
    #pragma unroll
    for (int j = 0; j < NK; ++j) {
        const v2f a = tp[j];               // Tv[m][k] = w[k-m], m = lane&15
        const int kx = 4 * j + koff;
        v2f b0 = { sH[0][kx][n15], sH[0][kx + 1][n15] };
        v2f b1 = { sH[1][kx][n15], sH[1][kx + 1][n15] };
        v2f b2 = { sH[2][kx][n15], sH[2][kx + 1][n15] };
        v2f b3 = { sH[3][kx][n15], sH[3][kx + 1][n15] };
        v2f b4 = { sH[4][kx][n15], sH[4][kx + 1][n15] };
        o0 = __builtin_amdgcn_wmma_f32_16x16x4_f32(false, a, false, b0, (short)0, o0, false, false);
        o1 = __builtin_amdgcn_wmma_f32_16x16x4_f32(false, a, false, b1, (short)0, o1, false, false);
        o2 = __builtin_amdgcn_wmma_f32_16x16x4_f32(false, a, false, b2, (short)0, o2, false, false);
        o3 = __builtin_amdgcn_wmma_f32_16x16x4_f32(false, a, false, b3, (short)0, o3, false, false);
        o4 = __builtin_amdgcn_wmma_f32_16x16x4_f32(false, a, false, b4, (short)0, o4, false, false);
    }

    // ---- Phase 3: SSIM map + wave reduction (8 pixels per lane) ----
    const float C1 = 1.0e-4f;   // 0.01^2
    const float C2 = 9.0e-4f;   // 0.03^2
    float s = 0.0f;
    #pragma unroll
    for (int v = 0; v < 8; ++v) {
        float mu1 = o0[v], mu2 = o1[v];
        float ex2 = o2[v], ey2 = o3[v], exy = o4[v];
        float mu1s = mu1 * mu1, mu2s = mu2 * mu2, mu12 = mu1 * mu2;
        float s1 = ex2 - mu1s, s2 = ey2 - mu2s, s12 = exy - mu12;
        float num = (2.0f * mu12 + C1) * (2.0f * s12 + C2);
        float den = (mu1s + mu2s + C1) * (s1 + s2 + C2);
        s += num / den;
    }
    #pragma unroll
    for (int m = 16; m >= 1; m >>= 1) s += __shfl_xor(s, m, 32);
    if (lane == 0) {
        if (atomicMode) {
            atomicAdd(target, s);
        } else {
            int bid = (blockIdx.z * gridDim.y + blockIdx.y) * gridDim.x + blockIdx.x;
            target[bid] = s;
        }
    }
}

__global__ __launch_bounds__(256) void ssim_finalize_sums(
        const float* __restrict__ sums, int n, float* __restrict__ out, float invN)
{
    __shared__ float red[256];
    float s = 0.0f;
    for (int i = threadIdx.x; i < n; i += 256) s += sums[i];
    red[threadIdx.x] = s;
    __syncthreads();
    #pragma unroll
    for (int w = 128; w >= 32; w >>= 1) {
        if ((int)threadIdx.x < w) red[threadIdx.x] += red[threadIdx.x + w];
        __syncthreads();
    }
    if (threadIdx.x < 32) {
        float v = red[threadIdx.x];
        #pragma unroll
        for (int m = 16; m >= 1; m >>= 1) v += __shfl_xor(v, m, 32);
        if (threadIdx.x == 0) out[0] = 1.0f - v * invN;
    }
}

__global__ void ssim_zero_scalar(float* p) { if (threadIdx.x == 0) p[0] = 0.0f; }
__global__ void ssim_final_inplace(float* out, float invN) {
    out[0] = 1.0f - out[0] * invN;
}

extern "C" void kernel_launch(void* const* d_in, const int* in_sizes, int n_in,
                              void* d_out, int out_size, void* d_ws, size_t ws_size,
                              hipStream_t stream)
{
    const float* img1 = (const float*)d_in[0];
    const float* img2 = (const float*)d_in[1];
    float* out = (float*)d_out;
    float* ws  = (float*)d_ws;

    const dim3 grid(IMG_W / TILE, IMG_H / TILE, PLANES);
    const dim3 block(32, 1, 1);
    const int  nBlocks = (IMG_W / TILE) * (IMG_H / TILE) * PLANES;
    const float invN = 1.0f / (float)((size_t)PLANES * IMG_H * IMG_W);

    if (ws_size >= (size_t)nBlocks * sizeof(float)) {
        // Deterministic two-stage reduction through workspace.
        ssim_tile_wmma<<<grid, block, 0, stream>>>(img1, img2, ws, 0);
        ssim_finalize_sums<<<1, 256, 0, stream>>>(ws, nBlocks, out, invN);
    } else {
        // Fallback: atomic accumulation directly into d_out, then transform.
        ssim_zero_scalar<<<1, 32, 0, stream>>>(out);
        ssim_tile_wmma<<<grid, block, 0, stream>>>(img1, img2, out, 1);
        ssim_final_inplace<<<1, 1, 0, stream>>>(out, invN);
    }
}